// FEModel_36069135351897
// MI455X (gfx1250) — compile-verified
//
#include <hip/hip_runtime.h>
#include <hip/hip_bf16.h>

typedef __attribute__((ext_vector_type(2))) float v2f;
typedef __attribute__((ext_vector_type(8))) float v8f;

#define NEG_SLOPE 0.2f

// ---------- helpers: order-preserving float<->uint map for atomicMax ----------
__device__ __forceinline__ unsigned mapf(float f) {
    unsigned u = __float_as_uint(f);
    return (u >> 31) ? ~u : (u ^ 0x80000000u);
}
__device__ __forceinline__ float unmapf(unsigned m) {
    return (m >> 31) ? __uint_as_float(m ^ 0x80000000u) : __uint_as_float(~m);
}

// ---------- self-loop attr: mean of incoming edge_attr per dst ----------
__global__ void k_zero2(float* a, float* b, int n) {
    int i = blockIdx.x * blockDim.x + threadIdx.x;
    if (i < n) { a[i] = 0.f; b[i] = 0.f; }
}

__global__ void k_loop_accum(const int* __restrict__ ei, const float* __restrict__ ea,
                             float* sums, float* cnts, int E_) {
    int e = blockIdx.x * blockDim.x + threadIdx.x;
    if (e >= E_) return;
    int d = ei[E_ + e];
    atomicAdd(&sums[d], ea[e]);
    atomicAdd(&cnts[d], 1.0f);
}

__global__ void k_loop_div(float* sums, const float* cnts, int N_) {
    int i = blockIdx.x * blockDim.x + threadIdx.x;
    if (i < N_) sums[i] = sums[i] / fmaxf(cnts[i], 1.0f);
}

// ---------- x = emb[x_idx]  (float4-wide: 4 threads per 16-float row) ----------
__global__ void k_gather(const int* __restrict__ xidx, const float4* __restrict__ emb4,
                         float4* __restrict__ X4, int N_) {
    int i = blockIdx.x * blockDim.x + threadIdx.x;
    if (i >= N_ * 4) return;
    int n = i >> 2, j = i & 3;
    X4[i] = emb4[xidx[n] * 4 + j];
}

// ---------- h = X @ W  (M=16-row tiles per wave, N=32 via two 16x16 tiles) ----------
// Uses V_WMMA_F32_16X16X4_F32, K stepped by 4.  A: 16x4 f32 (2 VGPR), B: 4x16 f32 (2 VGPR).
template <int K>
__global__ void k_gemm_wmma(const float* __restrict__ X, const float* __restrict__ W,
                            float* __restrict__ Hout, int nrows) {
    int wave = (int)((blockIdx.x * blockDim.x + threadIdx.x) >> 5);
    int lane = threadIdx.x & 31;
    int m0 = wave * 16;
    if (m0 >= nrows) return;            // wave-uniform exit: EXEC all-ones inside WMMA
    int rc = lane & 15;                 // row (for A) / col (for B,C)
    int khalf = lane >> 4;              // selects K pair {0,1} vs {2,3}
    int grow = m0 + rc; if (grow >= nrows) grow = nrows - 1;   // clamp tail loads
    bool full = (m0 + 16 <= nrows);     // wave-uniform: no per-element store guards needed

#pragma unroll
    for (int nt = 0; nt < 2; ++nt) {
        v8f c = {};
#pragma unroll
        for (int k0 = 0; k0 < K; k0 += 4) {
            v2f a, b;
            a.x = X[grow * K + k0 + 2 * khalf];
            a.y = X[grow * K + k0 + 2 * khalf + 1];
            b.x = W[(k0 + 2 * khalf) * 32 + nt * 16 + rc];
            b.y = W[(k0 + 2 * khalf + 1) * 32 + nt * 16 + rc];
            c = __builtin_amdgcn_wmma_f32_16x16x4_f32(
                    /*neg_a=*/false, a, /*neg_b=*/false, b,
                    /*c_mod=*/(short)0, c, /*reuse_a=*/false, /*reuse_b=*/false);
        }
        if (full) {
#pragma unroll
            for (int r = 0; r < 8; ++r)
                Hout[(m0 + r + 8 * khalf) * 32 + nt * 16 + rc] = c[r];
        } else {
#pragma unroll
            for (int r = 0; r < 8; ++r) {
                int orow = m0 + r + 8 * khalf;
                if (orow < nrows) Hout[orow * 32 + nt * 16 + rc] = c[r];
            }
        }
    }
}

// ---------- hs = h @ a_s, hd = h @ a_d (one wave per node, shuffle reduce) ----------
__global__ void k_node_vec(const float* __restrict__ H, const float* __restrict__ as_,
                           const float* __restrict__ ad_, float* hs, float* hd, int N_) {
    int wave = (int)((blockIdx.x * blockDim.x + threadIdx.x) >> 5);
    int lane = threadIdx.x & 31;
    if (wave >= N_) return;
    float v = H[wave * 32 + lane];
    float s = v * as_[lane];
    float d = v * ad_[lane];
#pragma unroll
    for (int off = 16; off > 0; off >>= 1) {
        s += __shfl_xor(s, off, 32);
        d += __shfl_xor(d, off, 32);
    }
    if (lane == 0) { hs[wave] = s; hd[wave] = d; }
}

// ---------- ce = dot(We, ae)  (EDGE_DIM == 1 collapses e@a_e to ea * ce) ----------
__global__ void k_prep_ce(const float* __restrict__ We, const float* __restrict__ ae, float* ce) {
    int lane = threadIdx.x & 31;
    float v = We[lane] * ae[lane];
#pragma unroll
    for (int off = 16; off > 0; off >>= 1) v += __shfl_xor(v, off, 32);
    if (lane == 0) ce[0] = v;
}

// ---------- per-layer init: amax=-inf(mapped 0), denom=0, Out=b ----------
__global__ void k_layer_init(unsigned* amaxu, float* denom, float* Out,
                             const float* __restrict__ b, int N_) {
    int i = blockIdx.x * blockDim.x + threadIdx.x;
    if (i >= N_ * 32) return;
    Out[i] = b[i & 31];
    if (i < N_) { amaxu[i] = 0u; denom[i] = 0.f; }
}

// ---------- alpha = leaky_relu(hs[s]+hd[d]+ea*ce); atomicMax amax[d] ----------
__global__ void k_edge_alpha(const int* __restrict__ ei, const float* __restrict__ ea,
                             const float* __restrict__ loop_attr,
                             const float* __restrict__ hs, const float* __restrict__ hd,
                             const float* __restrict__ ce,
                             float* __restrict__ alpha, unsigned* amaxu, int E_, int N_) {
    int e = blockIdx.x * blockDim.x + threadIdx.x;
    if (e >= E_ + N_) return;
    int s, d; float av;
    if (e < E_) { s = ei[e]; d = ei[E_ + e]; av = ea[e]; }
    else        { s = e - E_; d = s;         av = loop_attr[s]; }
    float a = hs[s] + hd[d] + av * ce[0];
    a = (a > 0.f) ? a : a * NEG_SLOPE;
    alpha[e] = a;
    atomicMax(&amaxu[d], mapf(a));
}

// ---------- ex = exp(alpha - amax[d]) (in place); denom[d] += ex ----------
__global__ void k_edge_exp(const int* __restrict__ ei, float* __restrict__ alpha,
                           const unsigned* __restrict__ amaxu, float* denom, int E_, int N_) {
    int e = blockIdx.x * blockDim.x + threadIdx.x;
    if (e >= E_ + N_) return;
    int d = (e < E_) ? ei[E_ + e] : (e - E_);
    float ex = __expf(alpha[e] - unmapf(amaxu[d]));
    alpha[e] = ex;
    atomicAdd(&denom[d], ex);
}

// ---------- Out[d] += h[s] * (ex / (denom[d]+eps))  (one wave per edge, lane=channel) ----------
__global__ void k_edge_scatter(const int* __restrict__ ei, const float* __restrict__ exv,
                               const float* __restrict__ denom, const float* __restrict__ H,
                               float* Out, int E_, int N_) {
    int wave = (int)((blockIdx.x * blockDim.x + threadIdx.x) >> 5);
    int lane = threadIdx.x & 31;
    if (wave >= E_ + N_) return;
    int s, d;
    if (wave < E_) { s = ei[wave]; d = ei[E_ + wave]; }
    else           { s = wave - E_; d = s; }
    float coef = exv[wave] / (denom[d] + 1e-16f);
    atomicAdd(&Out[d * 32 + lane], H[s * 32 + lane] * coef);
}

__global__ void k_relu4(float4* t4, int n4) {
    int i = blockIdx.x * blockDim.x + threadIdx.x;
    if (i >= n4) return;
    float4 v = t4[i];
    v.x = fmaxf(v.x, 0.f); v.y = fmaxf(v.y, 0.f);
    v.z = fmaxf(v.z, 0.f); v.w = fmaxf(v.w, 0.f);
    t4[i] = v;
}

// ---------- final: out[n] = dot(t[n,:], Wl) + bl ----------
__global__ void k_final(const float* __restrict__ T, const float* __restrict__ Wl,
                        const float* __restrict__ bl, float* out, int N_) {
    int wave = (int)((blockIdx.x * blockDim.x + threadIdx.x) >> 5);
    int lane = threadIdx.x & 31;
    if (wave >= N_) return;
    float v = T[wave * 32 + lane] * Wl[lane];
#pragma unroll
    for (int off = 16; off > 0; off >>= 1) v += __shfl_xor(v, off, 32);
    if (lane == 0) out[wave] = v + bl[0];
}

extern "C" void kernel_launch(void* const* d_in, const int* in_sizes, int n_in,
                              void* d_out, int out_size, void* d_ws, size_t ws_size,
                              hipStream_t stream) {
    const int*   xidx = (const int*)d_in[0];
    const int*   ei   = (const int*)d_in[1];      // (2,E): [0..E)=src, [E..2E)=dst
    const float* ea   = (const float*)d_in[2];
    const float* emb  = (const float*)d_in[3];
    const float* W1   = (const float*)d_in[4];
    const float* as1  = (const float*)d_in[5];
    const float* ad1  = (const float*)d_in[6];
    const float* We1  = (const float*)d_in[7];
    const float* ae1  = (const float*)d_in[8];
    const float* b1   = (const float*)d_in[9];
    const float* W2   = (const float*)d_in[10];
    const float* as2  = (const float*)d_in[11];
    const float* ad2  = (const float*)d_in[12];
    const float* We2  = (const float*)d_in[13];
    const float* ae2  = (const float*)d_in[14];
    const float* b2   = (const float*)d_in[15];
    const float* Wl   = (const float*)d_in[16];
    const float* bl   = (const float*)d_in[17];
    float* out = (float*)d_out;

    const int N = in_sizes[0];       // x_idx: (N,1)
    const int E = in_sizes[2];       // edge_attr: (E,1)
    const int Etot = E + N;

    // workspace layout (floats); whole thing ~48 MB, fits the 192 MB L2
    float* ws   = (float*)d_ws;
    float* sums = ws;                 // N  -> becomes loop_attr
    float* cnts = ws + (size_t)N;     // N
    float* x1   = ws + (size_t)2*N;   // N*16
    float* h    = ws + (size_t)18*N;  // N*32
    float* t    = ws + (size_t)50*N;  // N*32 (layer out / next in; reused for layer2 out)
    float* hs   = ws + (size_t)82*N;  // N
    float* hd   = ws + (size_t)83*N;  // N
    unsigned* amaxu = (unsigned*)(ws + (size_t)84*N); // N
    float* denom = ws + (size_t)85*N; // N
    float* alpha = ws + (size_t)86*N; // Etot (alpha, then ex in place)
    float* ce    = ws + (size_t)86*N + (size_t)Etot;  // 1

    const int TB = 256;
    auto g = [&](long long jobs) { return (unsigned)((jobs + TB - 1) / TB); };
    const unsigned gN    = g(N);
    const unsigned gN32  = g((long long)N * 32);
    const unsigned gN8   = g((long long)N * 8);       // float4 over N*32
    const unsigned gE    = g(E);
    const unsigned gEt   = g(Etot);
    const unsigned gEtW  = g((long long)Etot * 32);   // wave per edge
    const unsigned gNW   = g((long long)N * 32);      // wave per node
    const unsigned gGemm = g((long long)((N + 15) / 16) * 32);

    // self-loop attr = mean of incoming edge_attr
    k_zero2<<<gN, TB, 0, stream>>>(sums, cnts, N);
    k_loop_accum<<<gE, TB, 0, stream>>>(ei, ea, sums, cnts, E);
    k_loop_div<<<gN, TB, 0, stream>>>(sums, cnts, N);

    // x = emb[x_idx]
    k_gather<<<g((long long)N * 4), TB, 0, stream>>>(xidx, (const float4*)emb,
                                                     (float4*)x1, N);

    // ---- layer 1 ----
    k_gemm_wmma<16><<<gGemm, TB, 0, stream>>>(x1, W1, h, N);
    k_node_vec<<<gNW, TB, 0, stream>>>(h, as1, ad1, hs, hd, N);
    k_prep_ce<<<1, 32, 0, stream>>>(We1, ae1, ce);
    k_layer_init<<<gN32, TB, 0, stream>>>(amaxu, denom, t, b1, N);
    k_edge_alpha<<<gEt, TB, 0, stream>>>(ei, ea, sums, hs, hd, ce, alpha, amaxu, E, N);
    k_edge_exp<<<gEt, TB, 0, stream>>>(ei, alpha, amaxu, denom, E, N);
    k_edge_scatter<<<gEtW, TB, 0, stream>>>(ei, alpha, denom, h, t, E, N);
    k_relu4<<<gN8, TB, 0, stream>>>((float4*)t, N * 8);

    // ---- layer 2 (h/hs/hd extracted from t before t is re-initialized to b2) ----
    k_gemm_wmma<32><<<gGemm, TB, 0, stream>>>(t, W2, h, N);
    k_node_vec<<<gNW, TB, 0, stream>>>(h, as2, ad2, hs, hd, N);
    k_prep_ce<<<1, 32, 0, stream>>>(We2, ae2, ce);
    k_layer_init<<<gN32, TB, 0, stream>>>(amaxu, denom, t, b2, N);
    k_edge_alpha<<<gEt, TB, 0, stream>>>(ei, ea, sums, hs, hd, ce, alpha, amaxu, E, N);
    k_edge_exp<<<gEt, TB, 0, stream>>>(ei, alpha, amaxu, denom, E, N);
    k_edge_scatter<<<gEtW, TB, 0, stream>>>(ei, alpha, denom, h, t, E, N);

    // final projection H=32 -> 1
    k_final<<<gNW, TB, 0, stream>>>(t, Wl, bl, out, N);
}